// GATLayers_15607911154173
// MI455X (gfx1250) — compile-verified
//
#include <hip/hip_runtime.h>
#include <math.h>

// ---------------------------------------------------------------------------
// GAT 2-layer pipeline for MI455X (gfx1250, wave32).
//   GEMMs via V_WMMA_F32_16X16X4_F32 (fp32-exact, matrix pipe), full tiles
//   only -> branch-free store epilogue; scalar tail kernel for row remainder.
//   Edge phase via native f32 global atomics (L2-resident buffers).
// ---------------------------------------------------------------------------

typedef float v2f __attribute__((ext_vector_type(2)));
typedef float v8f __attribute__((ext_vector_type(8)));

#define NEG_SLOPE 0.2f
#define BN_EPS 1e-5f

// ---------------- WMMA GEMM: C[n,OC] = A[n,K] @ B[K,OC] --------------------
// One 16x16 output tile per wave32; K stepped by 4 (V_WMMA_F32_16X16X4_F32).
// Covers only full 16-row tiles (nFull rows); EXEC all-ones throughout.
template <int K, int OC>
__global__ void gemm_wmma_f32(const float* __restrict__ A,
                              const float* __restrict__ B,
                              float* __restrict__ C, int nFull) {
  const int wavesPerBlock = blockDim.x >> 5;
  const int wave = blockIdx.x * wavesPerBlock + (threadIdx.x >> 5);
  const int lane = threadIdx.x & 31;
  const int tilesN = OC / 16;
  const int tilesM = nFull >> 4;
  if (wave >= tilesM * tilesN) return;  // wave-uniform, EXEC stays full
  const int tm = wave / tilesN;
  const int tn = wave % tilesN;

  // A 16x4 f32 layout: lane L -> M = L&15, VGPR j holds K = (L>>4)*2 + j
  const int rowA = tm * 16 + (lane & 15);
  const int kh = lane >> 4;  // K sub-slot (0/1)
  const int colB = tn * 16 + (lane & 15);

  v8f acc = {};
#pragma unroll 4
  for (int k0 = 0; k0 < K; k0 += 4) {
    const int ka = k0 + kh * 2;
    v2f a, b;
    a.x = A[(long)rowA * K + ka];
    a.y = A[(long)rowA * K + ka + 1];
    b.x = B[(long)ka * OC + colB];
    b.y = B[(long)(ka + 1) * OC + colB];
    acc = __builtin_amdgcn_wmma_f32_16x16x4_f32(
        /*neg_a=*/false, a, /*neg_b=*/false, b,
        /*c_mod=*/(short)0, acc, /*reuse_a=*/false, /*reuse_b=*/false);
  }
  // C/D layout: VGPR v, lane L -> M = v + (L>>4)*8, N = L&15
  const long base = (long)(tm * 16 + (lane >> 4) * 8) * OC + tn * 16 + (lane & 15);
#pragma unroll
  for (int v = 0; v < 8; ++v) C[base + (long)v * OC] = acc[v];
}

// ---------------- scalar GEMM tail for rows [nFull, nRows) -----------------
template <int K, int OC>
__global__ void gemm_tail(const float* __restrict__ A,
                          const float* __restrict__ B, float* __restrict__ C,
                          int nFull, int nRows) {
  const int t = blockIdx.x * blockDim.x + threadIdx.x;
  const int tail = nRows - nFull;
  if (t >= tail * OC) return;
  const int r = nFull + t / OC;
  const int c = t % OC;
  float acc = 0.f;
  for (int k = 0; k < K; ++k) acc += A[(long)r * K + k] * B[(long)k * OC + c];
  C[(long)r * OC + c] = acc;
}

// ---------------- wave32 sum reduction -------------------------------------
__device__ __forceinline__ float waveReduceSum(float v) {
#pragma unroll
  for (int off = 16; off > 0; off >>= 1) v += __shfl_xor(v, off, 32);
  return v;
}

// ---------------- attention scores: a_s/a_d [N,H] --------------------------
// One wave per (node, head): coalesced lane-strided loads + shuffle reduce.
__global__ void att_scores(const float* __restrict__ hx,
                           const float* __restrict__ att_src,
                           const float* __restrict__ att_dst,
                           float* __restrict__ a_s, float* __restrict__ a_d,
                           int n, int Cc) {  // H fixed = 2
  const int wave = blockIdx.x * (blockDim.x >> 5) + (threadIdx.x >> 5);
  const int lane = threadIdx.x & 31;
  if (wave >= n * 2) return;
  const int node = wave >> 1, h = wave & 1;
  const float* row = hx + (long)node * 2 * Cc + h * Cc;
  const float* as = att_src + h * Cc;
  const float* ad = att_dst + h * Cc;
  float ss = 0.f, sd = 0.f;
  for (int c = lane; c < Cc; c += 32) {
    const float v = row[c];
    ss += v * as[c];
    sd += v * ad[c];
  }
  ss = waveReduceSum(ss);
  sd = waveReduceSum(sd);
  if (lane == 0) {
    a_s[wave] = ss;
    a_d[wave] = sd;
  }
}

// ---------------- utility ---------------------------------------------------
__global__ void fill_kernel(float* __restrict__ p, float v, long n) {
  const long i = (long)blockIdx.x * blockDim.x + threadIdx.x;
  if (i < n) p[i] = v;
}

__device__ __forceinline__ void edge_sd(const int* __restrict__ ei, int e,
                                        int E_, int& s, int& d) {
  if (e < E_) { s = ei[e]; d = ei[E_ + e]; }
  else        { s = d = e - E_; }            // self loop
}

// sign-aware float atomic max via integer atomics (no CAS loop)
__device__ __forceinline__ void atomicMaxFloat(float* addr, float value) {
  if (value >= 0.f) atomicMax((int*)addr, __float_as_int(value));
  else              atomicMin((unsigned int*)addr, __float_as_uint(value));
}

// ---------------- edge pass 1: leaky-relu logits + segment max -------------
__global__ void edge_logits(const int* __restrict__ ei, int E_, int ET_,
                            const float* __restrict__ a_s,
                            const float* __restrict__ a_d,
                            float* __restrict__ p, float* __restrict__ m) {
  const int e = blockIdx.x * blockDim.x + threadIdx.x;
  if (e >= ET_) return;
  int s, d;
  edge_sd(ei, e, E_, s, d);
#pragma unroll
  for (int h = 0; h < 2; ++h) {
    float v = a_s[s * 2 + h] + a_d[d * 2 + h];
    v = v > 0.f ? v : NEG_SLOPE * v;
    p[e * 2 + h] = v;
    atomicMaxFloat(&m[d * 2 + h], v);
  }
}

// ---------------- edge pass 2: exp(e - m[dst]) + segment sum ---------------
__global__ void edge_exp(const int* __restrict__ ei, int E_, int ET_,
                         const float* __restrict__ m, float* __restrict__ p,
                         float* __restrict__ z) {
  const int e = blockIdx.x * blockDim.x + threadIdx.x;
  if (e >= ET_) return;
  int s, d;
  edge_sd(ei, e, E_, s, d);
#pragma unroll
  for (int h = 0; h < 2; ++h) {
    const float v = __expf(p[e * 2 + h] - m[d * 2 + h]);
    p[e * 2 + h] = v;
    unsafeAtomicAdd(&z[d * 2 + h], v);
  }
}

// ---------------- edge pass 3: alpha = p / z[dst] --------------------------
__global__ void edge_norm(const int* __restrict__ ei, int E_, int ET_,
                          const float* __restrict__ z, float* __restrict__ p) {
  const int e = blockIdx.x * blockDim.x + threadIdx.x;
  if (e >= ET_) return;
  int s, d;
  edge_sd(ei, e, E_, s, d);
#pragma unroll
  for (int h = 0; h < 2; ++h) p[e * 2 + h] /= z[d * 2 + h];
}

// ---------------- edge pass 4: out[dst] += alpha * hx[src] -----------------
// thread t -> (edge = t / HC, channel = t % HC): consecutive lanes touch
// consecutive channels of one edge => coalesced gather + cacheline-merged
// atomics into the L2-resident accumulator.
template <int HC, int Cc>
__global__ void gat_scatter(const int* __restrict__ ei, int E_, int ET_,
                            const float* __restrict__ hx,
                            const float* __restrict__ alpha,
                            float* __restrict__ out) {
  const long t = (long)blockIdx.x * blockDim.x + threadIdx.x;
  if (t >= (long)ET_ * HC) return;
  const int e = (int)(t / HC);
  const int c = (int)(t % HC);
  const int h = c / Cc;
  int s, d;
  edge_sd(ei, e, E_, s, d);
  const float v = hx[(long)s * HC + c] * alpha[e * 2 + h];
  unsafeAtomicAdd(&out[(long)d * HC + c], v);
}

// ---------------- BN stats: per-column sum / sumsq (coalesced) -------------
template <int OC>
__global__ void bn_stats(const float* __restrict__ X,
                         const float* __restrict__ bias,
                         float* __restrict__ csum, float* __restrict__ csq,
                         long total) {
  __shared__ float ssum[OC];
  __shared__ float ssq[OC];
  const int t = threadIdx.x;
  for (int c = t; c < OC; c += blockDim.x) { ssum[c] = 0.f; ssq[c] = 0.f; }
  __syncthreads();
  const int ITER = 16;
  const long base = (long)blockIdx.x * blockDim.x * ITER;
  for (int i = 0; i < ITER; ++i) {
    const long idx = base + (long)i * blockDim.x + t;
    if (idx < total) {
      const int c = (int)(idx & (OC - 1));
      const float v = X[idx] + bias[c];
      atomicAdd(&ssum[c], v);          // DS_ADD_F32
      atomicAdd(&ssq[c], v * v);
    }
  }
  __syncthreads();
  for (int c = t; c < OC; c += blockDim.x) {
    unsafeAtomicAdd(&csum[c], ssum[c]);
    unsafeAtomicAdd(&csq[c], ssq[c]);
  }
}

// ---------------- BN apply + relu (in place) -------------------------------
template <int OC>
__global__ void bn_apply(float* __restrict__ X, const float* __restrict__ bias,
                         const float* __restrict__ gamma,
                         const float* __restrict__ beta,
                         const float* __restrict__ csum,
                         const float* __restrict__ csq, long total,
                         float invN) {
  const long idx = (long)blockIdx.x * blockDim.x + threadIdx.x;
  if (idx >= total) return;
  const int c = (int)(idx & (OC - 1));
  const float mu = csum[c] * invN;
  const float var = csq[c] * invN - mu * mu;
  float v = X[idx] + bias[c];
  v = (v - mu) * rsqrtf(var + BN_EPS) * gamma[c] + beta[c];
  X[idx] = v > 0.f ? v : 0.f;
}

// ---------------------------------------------------------------------------
extern "C" void kernel_launch(void* const* d_in, const int* in_sizes, int n_in,
                              void* d_out, int out_size, void* d_ws,
                              size_t ws_size, hipStream_t stream) {
  const float* x        = (const float*)d_in[0];
  const int*   ei       = (const int*)d_in[1];   // [2,E] (src row, dst row)
  const float* W1       = (const float*)d_in[2];
  const float* att_src1 = (const float*)d_in[3];
  const float* att_dst1 = (const float*)d_in[4];
  const float* bias1    = (const float*)d_in[5];
  const float* gamma1   = (const float*)d_in[6];
  const float* beta1    = (const float*)d_in[7];
  const float* W2       = (const float*)d_in[8];
  const float* att_src2 = (const float*)d_in[9];
  const float* att_dst2 = (const float*)d_in[10];
  const float* bias2    = (const float*)d_in[11];
  const float* gamma2   = (const float*)d_in[12];
  const float* beta2    = (const float*)d_in[13];
  float* out2 = (float*)d_out;

  const int F = 64;
  const int N = in_sizes[0] / F;        // 20000
  const int E = in_sizes[1] / 2;        // 640000
  const int ET = E + N;                 // edges + self loops
  const int NF = N & ~15;               // rows covered by full WMMA tiles
  const int HC1 = 128, C1 = 64;
  const int HC2 = 256, C2 = 128;

  // workspace carve-up (floats)
  float* ws = (float*)d_ws;
  size_t off = 0;
  float* hx1 = ws + off; off += (size_t)N * HC1;   // transformed feats, L1
  float* h1  = ws + off; off += (size_t)N * HC1;   // aggregate -> BN -> relu
  float* hx2 = ws + off; off += (size_t)N * HC2;   // transformed feats, L2
  float* a_s = ws + off; off += (size_t)N * 2;
  float* a_d = ws + off; off += (size_t)N * 2;
  float* m   = ws + off; off += (size_t)N * 2;     // segment max
  float* z   = ws + off; off += (size_t)N * 2;     // segment sum
  float* p   = ws + off; off += (size_t)ET * 2;    // logits -> exp -> alpha
  float* csum = ws + off; off += 256;
  float* csq  = ws + off; off += 256;
  (void)ws_size; (void)n_in; (void)out_size;

  const int B = 256;
  auto blocksFor = [](long n, int b) { return (int)((n + b - 1) / b); };
  const int ebl = blocksFor(ET, B);
  const int tail = N - NF;

  // ======================== Layer 1 ========================
  {
    const int tiles = (NF >> 4) * (HC1 / 16);
    gemm_wmma_f32<64, 128><<<blocksFor(tiles, 8), 256, 0, stream>>>(x, W1, hx1, NF);
    if (tail)
      gemm_tail<64, 128><<<blocksFor((long)tail * HC1, B), B, 0, stream>>>(x, W1,
                                                                           hx1, NF, N);
  }
  att_scores<<<blocksFor((long)N * 2, 8), B, 0, stream>>>(hx1, att_src1, att_dst1,
                                                          a_s, a_d, N, C1);
  fill_kernel<<<blocksFor((long)N * 2, B), B, 0, stream>>>(m, -INFINITY, (long)N * 2);
  fill_kernel<<<blocksFor((long)N * 2, B), B, 0, stream>>>(z, 0.f, (long)N * 2);
  fill_kernel<<<blocksFor((long)N * HC1, B), B, 0, stream>>>(h1, 0.f, (long)N * HC1);
  fill_kernel<<<1, B, 0, stream>>>(csum, 0.f, 256);
  fill_kernel<<<1, B, 0, stream>>>(csq, 0.f, 256);

  edge_logits<<<ebl, B, 0, stream>>>(ei, E, ET, a_s, a_d, p, m);
  edge_exp<<<ebl, B, 0, stream>>>(ei, E, ET, m, p, z);
  edge_norm<<<ebl, B, 0, stream>>>(ei, E, ET, z, p);
  gat_scatter<128, 64><<<blocksFor((long)ET * HC1, B), B, 0, stream>>>(ei, E, ET,
                                                                       hx1, p, h1);
  {
    const long total = (long)N * HC1;
    bn_stats<128><<<blocksFor(total, B * 16), B, 0, stream>>>(h1, bias1, csum, csq, total);
    bn_apply<128><<<blocksFor(total, B), B, 0, stream>>>(h1, bias1, gamma1, beta1,
                                                         csum, csq, total,
                                                         1.f / (float)N);
  }

  // ======================== Layer 2 ========================
  {
    const int tiles = (NF >> 4) * (HC2 / 16);
    gemm_wmma_f32<128, 256><<<blocksFor(tiles, 8), 256, 0, stream>>>(h1, W2, hx2, NF);
    if (tail)
      gemm_tail<128, 256><<<blocksFor((long)tail * HC2, B), B, 0, stream>>>(h1, W2,
                                                                            hx2, NF, N);
  }
  att_scores<<<blocksFor((long)N * 2, 8), B, 0, stream>>>(hx2, att_src2, att_dst2,
                                                          a_s, a_d, N, C2);
  fill_kernel<<<blocksFor((long)N * 2, B), B, 0, stream>>>(m, -INFINITY, (long)N * 2);
  fill_kernel<<<blocksFor((long)N * 2, B), B, 0, stream>>>(z, 0.f, (long)N * 2);
  fill_kernel<<<blocksFor((long)N * HC2, B), B, 0, stream>>>(out2, 0.f, (long)N * HC2);
  fill_kernel<<<1, B, 0, stream>>>(csum, 0.f, 256);
  fill_kernel<<<1, B, 0, stream>>>(csq, 0.f, 256);

  edge_logits<<<ebl, B, 0, stream>>>(ei, E, ET, a_s, a_d, p, m);
  edge_exp<<<ebl, B, 0, stream>>>(ei, E, ET, m, p, z);
  edge_norm<<<ebl, B, 0, stream>>>(ei, E, ET, z, p);
  gat_scatter<256, 128><<<blocksFor((long)ET * HC2, B), B, 0, stream>>>(ei, E, ET,
                                                                        hx2, p, out2);
  {
    const long total = (long)N * HC2;
    bn_stats<256><<<blocksFor(total, B * 16), B, 0, stream>>>(out2, bias2, csum, csq, total);
    bn_apply<256><<<blocksFor(total, B), B, 0, stream>>>(out2, bias2, gamma2, beta2,
                                                         csum, csq, total,
                                                         1.f / (float)N);
  }
}